// FastGCNv2_42691974922775
// MI455X (gfx1250) — compile-verified
//
#include <hip/hip_runtime.h>
#include <hip/hip_bf16.h>

// ---------------------------------------------------------------------------
// FastGCNv2 on MI455X (gfx1250, wave32):
//   pre = A@x ; h = relu(bn(pre@W0+b0)) ; h = A@h ; h = relu(bn(h@W1+b1))
//   h = A@h ; out = log_softmax(h@W2+b2)
// Dense GEMMs use V_WMMA_F32_16X16X4_F32 (full f32 precision, 16x16 tile/wave).
// SpMM uses native f32 global atomics (features are L2-resident: 51MB << 192MB).
// ---------------------------------------------------------------------------

#define NNODES 100000
#define FDIM   128
#define HDIM   128
#define CDIM   47
#define NEDGES 1600000
#define BN_EPS 1e-5f
#define ATS    132   // padded LDS row stride (floats): 132 % 64 = 4 -> conflict-free

typedef __attribute__((ext_vector_type(2))) float v2f;
typedef __attribute__((ext_vector_type(8))) float v8f;

// ---------------- zero fill ----------------
__global__ void gcn_zero(float* __restrict__ p, int n) {
  int i = blockIdx.x * blockDim.x + threadIdx.x;
  if (i < n) p[i] = 0.0f;
}

// ---------------- SpMM: out[r,:] += val * h[c,:]  (one wave per edge) -------
__global__ void gcn_spmm(const int* __restrict__ rows, const int* __restrict__ cols,
                         const float* __restrict__ vals, const float* __restrict__ h,
                         float* __restrict__ out) {
  int gid  = blockIdx.x * blockDim.x + threadIdx.x;
  int e    = gid >> 5;
  int lane = gid & 31;
  if (e >= NEDGES) return;
  int   r = rows[e];
  int   c = cols[e];
  float v = vals[e];
  float4 d = reinterpret_cast<const float4*>(h + (size_t)c * HDIM)[lane];
  float* o = out + (size_t)r * HDIM + lane * 4;
  unsafeAtomicAdd(o + 0, v * d.x);
  unsafeAtomicAdd(o + 1, v * d.y);
  unsafeAtomicAdd(o + 2, v * d.z);
  unsafeAtomicAdd(o + 3, v * d.w);
}

// ---------------- GEMM (N x 128) @ (128 x 128) + bias, f32 WMMA ------------
// 256 threads = 8 waves; wave w computes the 16x16 tile at cols [16w,16w+16).
__global__ void gcn_gemm128(const float* __restrict__ A, const float* __restrict__ W,
                            const float* __restrict__ bias, float* __restrict__ out) {
  __shared__ float Atile[16 * ATS];
  const int rowbase = blockIdx.x * 16;
  const int tid = threadIdx.x;

  // stage 16x128 A tile (512 float4), padded rows for bank-conflict-free K reads
  {
    const float4* src = reinterpret_cast<const float4*>(A + (size_t)rowbase * HDIM);
    float4* dst = reinterpret_cast<float4*>(Atile);
    int i0 = tid;                 // 0..255
    dst[(i0 >> 5) * (ATS / 4) + (i0 & 31)] = src[i0];
    int i1 = tid + 256;           // 256..511
    dst[(i1 >> 5) * (ATS / 4) + (i1 & 31)] = src[i1];
  }
  __syncthreads();

  const int wave  = tid >> 5;
  const int lane  = tid & 31;
  const int nbase = wave * 16;
  const int m     = lane & 15;            // A row / C column index within tile
  const int koff  = (lane < 16) ? 0 : 2;  // ISA 16x4 f32 A layout

  v8f acc = {};
#pragma unroll
  for (int k0 = 0; k0 < HDIM; k0 += 4) {
    v2f a, b;
    a.x = Atile[m * ATS + k0 + koff];
    a.y = Atile[m * ATS + k0 + koff + 1];
    b.x = W[(k0 + koff) * HDIM + nbase + m];
    b.y = W[(k0 + koff + 1) * HDIM + nbase + m];
    acc = __builtin_amdgcn_wmma_f32_16x16x4_f32(false, a, false, b, (short)0, acc,
                                                false, false);
  }

  const int col = nbase + m;
  const float bv = bias[col];
#pragma unroll
  for (int r = 0; r < 8; ++r) {
    int row = rowbase + r + ((lane >= 16) ? 8 : 0);
    out[(size_t)row * HDIM + col] = acc[r] + bv;
  }
}

// ---------------- final GEMM (N x 128) @ (128 x 47) + bias -----------------
// 96 threads = 3 waves cover the 47 (pad to 48) output columns.
__global__ void gcn_gemm_out(const float* __restrict__ A, const float* __restrict__ W2,
                             const float* __restrict__ b2, float* __restrict__ out) {
  __shared__ float Atile[16 * ATS];
  const int rowbase = blockIdx.x * 16;
  const int tid = threadIdx.x;

  {
    const float4* src = reinterpret_cast<const float4*>(A + (size_t)rowbase * HDIM);
    float4* dst = reinterpret_cast<float4*>(Atile);
    for (int i = tid; i < 512; i += 96)
      dst[(i >> 5) * (ATS / 4) + (i & 31)] = src[i];
  }
  __syncthreads();

  const int wave  = tid >> 5;
  const int lane  = tid & 31;
  const int nbase = wave * 16;
  const int m     = lane & 15;
  const int koff  = (lane < 16) ? 0 : 2;
  const int col   = nbase + m;
  const bool valid = (col < CDIM);
  const int ccol  = valid ? col : (CDIM - 1);
  const float sel = valid ? 1.0f : 0.0f;   // zero-pad column 47 without branching

  v8f acc = {};
#pragma unroll
  for (int k0 = 0; k0 < HDIM; k0 += 4) {
    v2f a, b;
    a.x = Atile[m * ATS + k0 + koff];
    a.y = Atile[m * ATS + k0 + koff + 1];
    b.x = W2[(k0 + koff) * CDIM + ccol] * sel;
    b.y = W2[(k0 + koff + 1) * CDIM + ccol] * sel;
    acc = __builtin_amdgcn_wmma_f32_16x16x4_f32(false, a, false, b, (short)0, acc,
                                                false, false);
  }

  if (valid) {
    const float bv = b2[col];
#pragma unroll
    for (int r = 0; r < 8; ++r) {
      int row = rowbase + r + ((lane >= 16) ? 8 : 0);
      out[(size_t)row * CDIM + col] = acc[r] + bv;
    }
  }
}

// ---------------- per-column sum / sum-of-squares (for BatchNorm) ----------
__global__ void gcn_colstats(const float* __restrict__ z, float* __restrict__ stats) {
  const int c = threadIdx.x & 127;
  const int rpt = blockDim.x >> 7;                 // rows handled per block pass
  int r = blockIdx.x * rpt + (threadIdx.x >> 7);
  const int rstride = gridDim.x * rpt;
  float s = 0.0f, sq = 0.0f;
  for (; r < NNODES; r += rstride) {
    float v = z[(size_t)r * HDIM + c];
    s += v; sq += v * v;
  }
  unsafeAtomicAdd(&stats[c], s);
  unsafeAtomicAdd(&stats[HDIM + c], sq);
}

// ---------------- BatchNorm(train, biased var) + ReLU, in place ------------
__global__ void gcn_bnrelu(float* __restrict__ z, const float* __restrict__ stats,
                           const float* __restrict__ g, const float* __restrict__ b) {
  int idx = blockIdx.x * blockDim.x + threadIdx.x;
  if (idx >= NNODES * HDIM) return;
  int c = idx & 127;
  const float inv_n = 1.0f / (float)NNODES;
  float m   = stats[c] * inv_n;
  float var = stats[HDIM + c] * inv_n - m * m;
  float y = (z[idx] - m) * rsqrtf(var + BN_EPS) * g[c] + b[c];
  z[idx] = fmaxf(y, 0.0f);
}

// ---------------- row-wise log_softmax over 47 classes (wave per row) ------
__global__ void gcn_logsoftmax(float* __restrict__ z) {
  int gid  = blockIdx.x * blockDim.x + threadIdx.x;
  int row  = gid >> 5;
  int lane = gid & 31;
  if (row >= NNODES) return;
  float* p = z + (size_t)row * CDIM;
  float v0 = p[lane];                                   // lanes 0..31 valid
  float v1 = (lane < CDIM - 32) ? p[lane + 32] : -INFINITY;
  float mx = fmaxf(v0, v1);
#pragma unroll
  for (int off = 16; off; off >>= 1) mx = fmaxf(mx, __shfl_xor(mx, off, 32));
  float s = expf(v0 - mx) + ((lane < CDIM - 32) ? expf(v1 - mx) : 0.0f);
#pragma unroll
  for (int off = 16; off; off >>= 1) s += __shfl_xor(s, off, 32);
  float lse = mx + logf(s);
  p[lane] = v0 - lse;
  if (lane < CDIM - 32) p[lane + 32] = v1 - lse;
}

// ---------------------------------------------------------------------------
extern "C" void kernel_launch(void* const* d_in, const int* in_sizes, int n_in,
                              void* d_out, int out_size, void* d_ws, size_t ws_size,
                              hipStream_t stream) {
  const float* x   = (const float*)d_in[0];
  const float* ev  = (const float*)d_in[1];
  const float* W0  = (const float*)d_in[2];
  const float* b0  = (const float*)d_in[3];
  const float* g0  = (const float*)d_in[4];
  const float* be0 = (const float*)d_in[5];
  const float* W1  = (const float*)d_in[6];
  const float* b1  = (const float*)d_in[7];
  const float* g1  = (const float*)d_in[8];
  const float* be1 = (const float*)d_in[9];
  const float* W2  = (const float*)d_in[10];
  const float* b2  = (const float*)d_in[11];
  const int* erow  = (const int*)d_in[12];
  const int* ecol  = (const int*)d_in[13];
  float* out = (float*)d_out;

  float* bufA  = (float*)d_ws;            // N x 128 (spmm accumulator)
  float* bufB  = bufA + (size_t)NNODES * HDIM;  // N x 128 (dense layer output)
  float* stats = bufB + (size_t)NNODES * HDIM;  // 256 floats: colsum | colsumsq

  const int NH = NNODES * HDIM;
  dim3 blk256(256);
  dim3 zgrid((NH + 255) / 256);
  dim3 sgrid(NEDGES / 8);                 // 8 edge-waves per 256-thread block
  dim3 ggrid(NNODES / 16);                // 6250 row tiles
  dim3 lgrid(NNODES / 8);                 // 8 row-waves per block

  // ----- pre = A @ x -----
  gcn_zero<<<zgrid, blk256, 0, stream>>>(bufA, NH);
  gcn_spmm<<<sgrid, blk256, 0, stream>>>(erow, ecol, ev, x, bufA);

  // ----- layer 0: relu(bn(pre @ W0 + b0)) -----
  gcn_gemm128<<<ggrid, blk256, 0, stream>>>(bufA, W0, b0, bufB);
  gcn_zero<<<dim3(1), blk256, 0, stream>>>(stats, 2 * HDIM);
  gcn_colstats<<<dim3(1024), blk256, 0, stream>>>(bufB, stats);
  gcn_bnrelu<<<zgrid, blk256, 0, stream>>>(bufB, stats, g0, be0);

  // ----- middle layer: relu(bn((A @ h) @ W1 + b1)) -----
  gcn_zero<<<zgrid, blk256, 0, stream>>>(bufA, NH);
  gcn_spmm<<<sgrid, blk256, 0, stream>>>(erow, ecol, ev, bufB, bufA);
  gcn_gemm128<<<ggrid, blk256, 0, stream>>>(bufA, W1, b1, bufB);
  gcn_zero<<<dim3(1), blk256, 0, stream>>>(stats, 2 * HDIM);
  gcn_colstats<<<dim3(1024), blk256, 0, stream>>>(bufB, stats);
  gcn_bnrelu<<<zgrid, blk256, 0, stream>>>(bufB, stats, g1, be1);

  // ----- final layer: log_softmax((A @ h) @ W2 + b2) -----
  gcn_zero<<<zgrid, blk256, 0, stream>>>(bufA, NH);
  gcn_spmm<<<sgrid, blk256, 0, stream>>>(erow, ecol, ev, bufB, bufA);
  gcn_gemm_out<<<ggrid, dim3(96), 0, stream>>>(bufA, W2, b2, out);
  gcn_logsoftmax<<<lgrid, blk256, 0, stream>>>(out);
}